// HoCN_13469017440664
// MI455X (gfx1250) — compile-verified
//
#include <hip/hip_runtime.h>
#include <stdint.h>

// ---------------------------------------------------------------------------
// HoCN pipeline for MI455X (gfx1250), memory-bound sparse workload.
//  k_count    : degree histogram (int atomics)
//  k_scanblk / k_scansums / k_finalize : exclusive scan -> CSR offsets, dinv
//  k_fill     : CSR column fill via atomic cursors
//  k_spmm     : gather SpMM, wave32 per row, float4 per lane,
//               async global->LDS double buffering (gfx1250 async path)
//  k_query    : gather + elementwise products for the 8 outputs
// ---------------------------------------------------------------------------

#define TB 256
#define WPB (TB / 32)

#if defined(__gfx1250__)
#  define HOCN_ASYNC 1
#endif

#if defined(HOCN_ASYNC)

// ASYNCcnt wait: builtin exists on this toolchain (verified round 1);
// fall back to raw SOPP asm if a future toolchain lacks it.
#  if defined(__has_builtin)
#    if __has_builtin(__builtin_amdgcn_s_wait_asynccnt)
#      define HOCN_WAIT_ASYNC(n) __builtin_amdgcn_s_wait_asynccnt(n)
#    endif
#  endif
#  ifndef HOCN_WAIT_ASYNC
#    define HOCN_WAIT_ASYNC(n) asm volatile("s_wait_asynccnt " #n ::: "memory")
#  endif

// Per-lane async gather of 16B from global into LDS (VDST carries the LDS
// byte offset, VADDR the 64-bit global address; GV mode, tracked by ASYNCcnt).
__device__ __forceinline__ void hocn_async_b128(const void* gsrc, void* lds) {
    unsigned lds_off =
        (unsigned)(size_t)(__attribute__((address_space(3))) void*)lds;
    unsigned long long gaddr = (unsigned long long)(size_t)gsrc;
    asm volatile("global_load_async_to_lds_b128 %0, %1, off"
                 :
                 : "v"(lds_off), "v"(gaddr)
                 : "memory");
}
#endif  // HOCN_ASYNC

__device__ __forceinline__ float4 f4mul(float4 a, float4 b) {
    return make_float4(a.x * b.x, a.y * b.y, a.z * b.z, a.w * b.w);
}

// --------------------------- degree histogram ------------------------------
__global__ void k_count(const int* __restrict__ ei_row, int* __restrict__ cnt, int E) {
    int e = blockIdx.x * blockDim.x + threadIdx.x;
    if (e < E) atomicAdd(&cnt[ei_row[e]], 1);
}

// --------------------------- two-level scan --------------------------------
__global__ void k_scanblk(const int* __restrict__ cnt, int* __restrict__ excl,
                          int* __restrict__ bsums, int N) {
    __shared__ int s[256];
    int tid = threadIdx.x;
    int i = blockIdx.x * 256 + tid;
    int v = (i < N) ? cnt[i] : 0;
    s[tid] = v;
    __syncthreads();
    for (int d = 1; d < 256; d <<= 1) {
        int t = (tid >= d) ? s[tid - d] : 0;
        __syncthreads();
        s[tid] += t;
        __syncthreads();
    }
    if (i < N) excl[i] = s[tid] - v;
    if (tid == 255) bsums[blockIdx.x] = s[255];
}

__global__ void k_scansums(int* __restrict__ bsums, int nb) {
    __shared__ int s[256];
    __shared__ int carry;
    int tid = threadIdx.x;
    if (tid == 0) carry = 0;
    __syncthreads();
    for (int base = 0; base < nb; base += 256) {
        int i = base + tid;
        int v = (i < nb) ? bsums[i] : 0;
        s[tid] = v;
        __syncthreads();
        for (int d = 1; d < 256; d <<= 1) {
            int t = (tid >= d) ? s[tid - d] : 0;
            __syncthreads();
            s[tid] += t;
            __syncthreads();
        }
        if (i < nb) bsums[i] = s[tid] - v + carry;  // exclusive + running carry
        __syncthreads();
        if (tid == 0) carry += s[255];
        __syncthreads();
    }
}

__global__ void k_finalize(const int* __restrict__ cnt, const int* __restrict__ excl,
                           const int* __restrict__ bsums, int* __restrict__ offs,
                           int* __restrict__ cursor, float* __restrict__ degf,
                           float* __restrict__ dinv, int N, int E) {
    int i = blockIdx.x * blockDim.x + threadIdx.x;
    if (i < N) {
        int o = excl[i] + bsums[i >> 8];
        offs[i] = o;
        cursor[i] = o;
        int c = cnt[i];
        degf[i] = (float)c;
        dinv[i] = (c > 0) ? rsqrtf((float)c) : 0.0f;
    }
    if (i == 0) offs[N] = E;
}

// --------------------------- CSR fill --------------------------------------
__global__ void k_fill(const int* __restrict__ ei, int* __restrict__ cursor,
                       int* __restrict__ colc, int E) {
    int e = blockIdx.x * blockDim.x + threadIdx.x;
    if (e < E) {
        int r = ei[e];
        int c = ei[E + e];
        int pos = atomicAdd(&cursor[r], 1);
        colc[pos] = c;
    }
}

// --------------------------- gather SpMM -----------------------------------
// dst[r] = sum_{c in N(r)} dinv[r]*dinv[c] * src[c]; one wave32 per row,
// each lane owns 4 of 128 features.
__global__ void k_spmm(const int* __restrict__ offs, const int* __restrict__ colc,
                       const float* __restrict__ dinv, const float* __restrict__ src,
                       float* __restrict__ dst, int N) {
    int lane = threadIdx.x & 31;
    int winb = threadIdx.x >> 5;
    int row = blockIdx.x * WPB + winb;
    if (row >= N) return;

    int start = offs[row];
    int end = offs[row + 1];
    float dr = dinv[row];
    float4 acc = make_float4(0.f, 0.f, 0.f, 0.f);

#if defined(HOCN_ASYNC)
    // Double-buffered async global->LDS staging of neighbor feature rows:
    // the async engine gathers the next row while we FMA the current one.
    __shared__ float stage[WPB][2][128];
    int j = start;
    int buf = 0;
    int c_cur = 0;
    if (j < end) {
        c_cur = colc[j];
        hocn_async_b128(src + (size_t)c_cur * 128 + lane * 4,
                        &stage[winb][0][lane * 4]);
    }
    for (; j < end; ++j) {
        int c_nxt = 0;
        if (j + 1 < end) {
            c_nxt = colc[j + 1];
            hocn_async_b128(src + (size_t)c_nxt * 128 + lane * 4,
                            &stage[winb][buf ^ 1][lane * 4]);
            HOCN_WAIT_ASYNC(1);  // oldest (current buf) complete
        } else {
            HOCN_WAIT_ASYNC(0);
        }
        float v = dr * dinv[c_cur];
        float4 f = ((const float4*)&stage[winb][buf][0])[lane];
        acc.x += v * f.x;
        acc.y += v * f.y;
        acc.z += v * f.z;
        acc.w += v * f.w;
        buf ^= 1;
        c_cur = c_nxt;
    }
#else
    for (int j = start; j < end; ++j) {
        int c = colc[j];
        if (j + 2 < end) {
            int cp = colc[j + 2];
            __builtin_prefetch(src + (size_t)cp * 128 + lane * 4, 0, 0);
        }
        float v = dr * dinv[c];
        float4 f = ((const float4*)(src + (size_t)c * 128))[lane];
        acc.x += v * f.x;
        acc.y += v * f.y;
        acc.z += v * f.z;
        acc.w += v * f.w;
    }
#endif
    ((float4*)(dst + (size_t)row * 128))[lane] = acc;
}

// --------------------------- query stage -----------------------------------
__global__ void k_query(const int* __restrict__ eq, const float* __restrict__ x,
                        const float* __restrict__ h1, const float* __restrict__ h2,
                        const float* __restrict__ degf, float* __restrict__ out, int Q) {
    int lane = threadIdx.x & 31;
    int q = blockIdx.x * WPB + (threadIdx.x >> 5);
    if (q >= Q) return;

    int u = eq[q];
    int v = eq[Q + q];

    float4 a1 = ((const float4*)(h1 + (size_t)u * 128))[lane];  // h1[u]
    float4 b1 = ((const float4*)(h1 + (size_t)v * 128))[lane];  // h1[v]
    float4 a2 = ((const float4*)(h2 + (size_t)u * 128))[lane];  // h2[u]
    float4 b2 = ((const float4*)(h2 + (size_t)v * 128))[lane];  // h2[v]
    float4 xu = ((const float4*)(x + (size_t)u * 128))[lane];
    float4 xv = ((const float4*)(x + (size_t)v * 128))[lane];
    float du = degf[u];
    float dv = degf[v];

    size_t QD = (size_t)Q * 128;
    size_t base = (size_t)q * 128 + (size_t)lane * 4;

    *(float4*)(out + 0 * QD + base) = f4mul(a1, b1);  // count_1_1
    *(float4*)(out + 1 * QD + base) = f4mul(a1, b2);  // count_1_2
    *(float4*)(out + 2 * QD + base) = f4mul(a2, b1);  // count_2_1
    float4 tu = make_float4(a2.x - du * xu.x, a2.y - du * xu.y,
                            a2.z - du * xu.z, a2.w - du * xu.w);
    float4 tv = make_float4(b2.x - dv * xv.x, b2.y - dv * xv.y,
                            b2.z - dv * xv.z, b2.w - dv * xv.w);
    *(float4*)(out + 3 * QD + base) = f4mul(tu, tv);  // count_2_2
    *(float4*)(out + 4 * QD + base) = f4mul(a1, a2);  // count_self_1_2
    *(float4*)(out + 5 * QD + base) = f4mul(b1, b2);  // count_self_2_1

    if (lane == 0) {
        out[6 * QD + q] = du;       // deg_u
        out[6 * QD + Q + q] = dv;   // deg_v
    }
}

// --------------------------- host launcher ---------------------------------
extern "C" void kernel_launch(void* const* d_in, const int* in_sizes, int n_in,
                              void* d_out, int out_size, void* d_ws, size_t ws_size,
                              hipStream_t stream) {
    const float* x = (const float*)d_in[0];
    const int* edges = (const int*)d_in[1];       // [2, Q]
    const int* ei = (const int*)d_in[2];          // [2, E]
    int N = in_sizes[0] / 128;
    int Q = in_sizes[1] / 2;
    int E = in_sizes[2] / 2;

    // workspace carve-out (256B-aligned)
    char* w = (char*)d_ws;
    size_t off = 0;
    auto carve = [&](size_t bytes) -> void* {
        off = (off + 255) & ~(size_t)255;
        void* p = w + off;
        off += bytes;
        return p;
    };
    int nb = (N + 255) / 256;
    int*   cnt    = (int*)carve((size_t)N * 4);
    int*   excl   = (int*)carve((size_t)N * 4);
    int*   bsums  = (int*)carve((size_t)nb * 4);
    int*   offs   = (int*)carve(((size_t)N + 1) * 4);
    int*   cursor = (int*)carve((size_t)N * 4);
    int*   colc   = (int*)carve((size_t)E * 4);
    float* degf   = (float*)carve((size_t)N * 4);
    float* dinv   = (float*)carve((size_t)N * 4);
    float* h1     = (float*)carve((size_t)N * 128 * 4);
    float* h2     = (float*)carve((size_t)N * 128 * 4);

    (void)hipMemsetAsync(cnt, 0, (size_t)N * 4, stream);

    k_count<<<(E + TB - 1) / TB, TB, 0, stream>>>(ei, cnt, E);
    k_scanblk<<<nb, 256, 0, stream>>>(cnt, excl, bsums, N);
    k_scansums<<<1, 256, 0, stream>>>(bsums, nb);
    k_finalize<<<(N + TB - 1) / TB, TB, 0, stream>>>(cnt, excl, bsums, offs, cursor,
                                                     degf, dinv, N, E);
    k_fill<<<(E + TB - 1) / TB, TB, 0, stream>>>(ei, cursor, colc, E);

    int gSp = (N + WPB - 1) / WPB;
    k_spmm<<<gSp, TB, 0, stream>>>(offs, colc, dinv, x, h1, N);   // one_hop
    k_spmm<<<gSp, TB, 0, stream>>>(offs, colc, dinv, h1, h2, N);  // two_hop

    int gQ = (Q + WPB - 1) / WPB;
    k_query<<<gQ, TB, 0, stream>>>(edges, x, h1, h2, degf, (float*)d_out, Q);
}